// CrystalGCN_4277787427213
// MI455X (gfx1250) — compile-verified
//
#include <hip/hip_runtime.h>
#include <hip/hip_bf16.h>

typedef __attribute__((ext_vector_type(2))) float v2f;
typedef __attribute__((ext_vector_type(8))) float v8f;

#define NN      60000
#define MM      12
#define ORIG_D  92
#define NBR_D   41
#define DD      64
#define HH      128
#define TWO_D   128
#define KC      169      // 2*D + NBR_D
#define KCP     172      // padded to multiple of 4
#define LDA_C   173      // odd stride for A tile (conv)
#define LDB_C   136      // 136 % 32 == 8 -> half-wave bank groups disjoint
#define LDA_E   97
#define LDB_E   72       // 72 % 32 == 8
#define NCRYS   1024
#define GEPS    1e-5f

__device__ __forceinline__ float softplusf(float v) {
    // stable log(1+exp(v)) = max(v,0) + log1p(exp(-|v|))
    return fmaxf(v, 0.f) + log1pf(expf(-fabsf(v)));
}
__device__ __forceinline__ float sigmoidf(float v) {
    return 1.f / (1.f + expf(-v));
}

// ---------------------------------------------------------------------------
// Embedding: x[N x 64] = atom_fea[N x 92] @ emb_w[92 x 64] + emb_b
// block = 128 threads (4 waves), tile 64 rows x 64 cols, K = 92 (23 wmma steps)
// ---------------------------------------------------------------------------
__global__ __launch_bounds__(128) void embed_kernel(
    const float* __restrict__ af, const float* __restrict__ ew,
    const float* __restrict__ eb, float* __restrict__ x)
{
    __shared__ float As[64 * LDA_E];
    __shared__ float Bs[ORIG_D * LDB_E];
    const int tid = threadIdx.x;
    const int R0  = blockIdx.x * 64;

    for (int i = tid; i < 64 * ORIG_D; i += 128) {
        int r = i / ORIG_D, c = i - r * ORIG_D;
        int rg = R0 + r;
        As[r * LDA_E + c] = (rg < NN) ? af[(size_t)rg * ORIG_D + c] : 0.f;
    }
    for (int i = tid; i < ORIG_D * DD; i += 128) {
        int k = i >> 6, c = i & 63;
        Bs[k * LDB_E + c] = ew[k * DD + c];
    }
    __syncthreads();

    const int wv   = tid >> 5;
    const int lane = tid & 31;
    const int half = lane >> 4;
    const int l16  = lane & 15;
    const int c0   = wv << 4;            // 4 waves cover cols 0..63

    v8f acc[4] = {};
    for (int k = 0; k < ORIG_D; k += 4) {
        v2f b;
        b.x = Bs[(k + 2 * half) * LDB_E + c0 + l16];
        b.y = Bs[(k + 2 * half + 1) * LDB_E + c0 + l16];
#pragma unroll
        for (int rt = 0; rt < 4; ++rt) {
            v2f a;
            a.x = As[(rt * 16 + l16) * LDA_E + k + 2 * half];
            a.y = As[(rt * 16 + l16) * LDA_E + k + 2 * half + 1];
            acc[rt] = __builtin_amdgcn_wmma_f32_16x16x4_f32(
                false, a, false, b, (short)0, acc[rt], false, false);
        }
    }
    const int col  = c0 + l16;
    const float bv = eb[col];
#pragma unroll
    for (int rt = 0; rt < 4; ++rt) {
#pragma unroll
        for (int v = 0; v < 8; ++v) {
            int row = R0 + rt * 16 + half * 8 + v;
            if (row < NN) x[(size_t)row * DD + col] = acc[rt][v] + bv;
        }
    }
}

// ---------------------------------------------------------------------------
// Conv GEMM core: block tile = 48 rows (exactly 4 atoms) x 128 cols, K = 169
// 256 threads = 8 waves; wave w owns cols [16w,16w+16), 3 row tiles.
// ---------------------------------------------------------------------------
__device__ __forceinline__ void conv_load_tiles(
    int tid, int R0,
    const float* __restrict__ x, const float* __restrict__ nbr_fea,
    const int* __restrict__ nbr_idx, const float* __restrict__ fcw,
    float* As, float* Bs)
{
    for (int i = tid; i < KCP * TWO_D; i += 256) {
        int k = i >> 7, c = i & 127;
        Bs[k * LDB_C + c] = (k < KC) ? fcw[k * TWO_D + c] : 0.f;
    }
    for (int i = tid; i < 48 * KCP; i += 256) {
        int r = i / KCP, c = i - r * KCP;
        int rg = R0 + r;
        int atom = rg / MM;
        float v;
        if (c < DD)           v = x[(size_t)atom * DD + c];
        else if (c < 2 * DD)  v = x[(size_t)nbr_idx[rg] * DD + (c - DD)];
        else if (c < KC)      v = nbr_fea[(size_t)rg * NBR_D + (c - 2 * DD)];
        else                  v = 0.f;
        As[r * LDA_C + c] = v;
    }
}

__device__ __forceinline__ void conv_gemm(
    const float* As, const float* Bs,
    int c0, int half, int l16, v8f acc[3])
{
    for (int k = 0; k < KCP; k += 4) {
        v2f b;
        b.x = Bs[(k + 2 * half) * LDB_C + c0 + l16];
        b.y = Bs[(k + 2 * half + 1) * LDB_C + c0 + l16];
#pragma unroll
        for (int rt = 0; rt < 3; ++rt) {
            v2f a;
            a.x = As[(rt * 16 + l16) * LDA_C + k + 2 * half];
            a.y = As[(rt * 16 + l16) * LDA_C + k + 2 * half + 1];
            acc[rt] = __builtin_amdgcn_wmma_f32_16x16x4_f32(
                false, a, false, b, (short)0, acc[rt], false, false);
        }
    }
}

// Pass 1: accumulate per-feature sum / sumsq of g = tot @ W + b  (for BN1)
__global__ __launch_bounds__(256) void conv_pass1(
    const float* __restrict__ x, const float* __restrict__ nbr_fea,
    const int* __restrict__ nbr_idx,
    const float* __restrict__ fcw, const float* __restrict__ fcb,
    float* __restrict__ stat1)   // [0:128) sum, [128:256) sumsq
{
    __shared__ float As[48 * LDA_C];
    __shared__ float Bs[KCP * LDB_C];
    __shared__ float ssum[TWO_D];
    __shared__ float ssq[TWO_D];
    const int tid = threadIdx.x;
    const int R0  = blockIdx.x * 48;

    conv_load_tiles(tid, R0, x, nbr_fea, nbr_idx, fcw, As, Bs);
    if (tid < TWO_D) { ssum[tid] = 0.f; ssq[tid] = 0.f; }
    __syncthreads();

    const int lane = tid & 31, half = lane >> 4, l16 = lane & 15;
    const int c0 = (tid >> 5) << 4;
    v8f acc[3] = {};
    conv_gemm(As, Bs, c0, half, l16, acc);

    const int col = c0 + l16;
    const float bias = fcb[col];
    float cs = 0.f, cq = 0.f;
#pragma unroll
    for (int rt = 0; rt < 3; ++rt)
#pragma unroll
        for (int v = 0; v < 8; ++v) {
            float e = acc[rt][v] + bias;
            cs += e; cq += e * e;
        }
    atomicAdd(&ssum[col], cs);
    atomicAdd(&ssq[col], cq);
    __syncthreads();
    if (tid < TWO_D) {
        atomicAdd(&stat1[tid], ssum[tid]);
        atomicAdd(&stat1[TWO_D + tid], ssq[tid]);
    }
}

// Pass 2: recompute g, apply BN1, gate = sigmoid(filt)*softplus(core),
// reduce over the 12 neighbors (4 atoms fully inside the block),
// write s[N x 64] and accumulate BN2 stats.
__global__ __launch_bounds__(256) void conv_pass2(
    const float* __restrict__ x, const float* __restrict__ nbr_fea,
    const int* __restrict__ nbr_idx,
    const float* __restrict__ fcw, const float* __restrict__ fcb,
    const float* __restrict__ bnp1,  // [0:128) mean, [128:256) invstd
    const float* __restrict__ g1, const float* __restrict__ b1,
    float* __restrict__ s_out, float* __restrict__ stat2) // [0:64) sum, [64:128) sumsq
{
    __shared__ float As[48 * LDA_C];          // reused as normalized-g buffer
    __shared__ float Bs[KCP * LDB_C];
    __shared__ float s2[DD];
    __shared__ float s2q[DD];
    const int tid = threadIdx.x;
    const int R0  = blockIdx.x * 48;

    conv_load_tiles(tid, R0, x, nbr_fea, nbr_idx, fcw, As, Bs);
    if (tid < DD) { s2[tid] = 0.f; s2q[tid] = 0.f; }
    __syncthreads();

    const int lane = tid & 31, half = lane >> 4, l16 = lane & 15;
    const int c0 = (tid >> 5) << 4;
    v8f acc[3] = {};
    conv_gemm(As, Bs, c0, half, l16, acc);

    const int col = c0 + l16;
    const float sc = bnp1[TWO_D + col] * g1[col];
    const float of = (fcb[col] - bnp1[col]) * sc + b1[col];

    __syncthreads();                           // done reading As -> reuse LDS
    float* gsm = As;                           // 48 rows x stride 132
#pragma unroll
    for (int rt = 0; rt < 3; ++rt)
#pragma unroll
        for (int v = 0; v < 8; ++v) {
            int row = rt * 16 + half * 8 + v;
            gsm[row * 132 + col] = acc[rt][v] * sc + of;
        }
    __syncthreads();

    const int a = tid >> 6, d = tid & 63;      // 4 atoms x 64 features
    float sum = 0.f;
#pragma unroll
    for (int m = 0; m < MM; ++m) {
        int row = a * MM + m;
        float f = gsm[row * 132 + d];
        float c = gsm[row * 132 + DD + d];
        sum += sigmoidf(f) * softplusf(c);
    }
    int atom = blockIdx.x * 4 + a;
    s_out[(size_t)atom * DD + d] = sum;
    atomicAdd(&s2[d], sum);
    atomicAdd(&s2q[d], sum * sum);
    __syncthreads();
    if (tid < DD) {
        atomicAdd(&stat2[tid], s2[tid]);
        atomicAdd(&stat2[DD + tid], s2q[tid]);
    }
}

// ---------------------------------------------------------------------------
__global__ void bn_finalize(const float* __restrict__ stat, float* __restrict__ bnp,
                            int n, float inv_cnt)
{
    int t = blockIdx.x * blockDim.x + threadIdx.x;
    if (t < n) {
        float m = stat[t] * inv_cnt;
        float v = stat[n + t] * inv_cnt - m * m;
        bnp[t] = m;
        bnp[n + t] = rsqrtf(v + GEPS);
    }
}

__global__ void update_x(float* __restrict__ x, const float* __restrict__ s,
                         const float* __restrict__ bnp2,
                         const float* __restrict__ g2, const float* __restrict__ b2)
{
    int i = blockIdx.x * blockDim.x + threadIdx.x;
    if (i < NN * DD) {
        int d = i & 63;
        float sv = (s[i] - bnp2[d]) * bnp2[DD + d] * g2[d] + b2[d];
        x[i] = softplusf(x[i] + sv);
    }
}

__global__ void zero_f32(float* __restrict__ p, int n)
{
    int i = blockIdx.x * blockDim.x + threadIdx.x;
    if (i < n) p[i] = 0.f;
}

__global__ void pool_kernel(const float* __restrict__ x, const int* __restrict__ cid,
                            float* __restrict__ crys, float* __restrict__ cnt)
{
    int idx = blockIdx.x * blockDim.x + threadIdx.x;
    if (idx < NN * DD) {
        int i = idx >> 6, d = idx & 63;
        int c = cid[i];
        atomicAdd(&crys[(size_t)c * DD + d], x[idx]);
        if (d == 0) atomicAdd(&cnt[c], 1.0f);
    }
}

__global__ __launch_bounds__(128) void head_kernel(
    const float* __restrict__ crys, const float* __restrict__ cnt,
    const float* __restrict__ cfw, const float* __restrict__ cfb,
    const float* __restrict__ ow, const float* __restrict__ ob,
    float* __restrict__ out)
{
    __shared__ float p[DD];
    __shared__ float red[HH];
    const int b = blockIdx.x, t = threadIdx.x;
    if (t < DD) {
        float c = cnt[b];
        c = c > 1.f ? c : 1.f;
        p[t] = softplusf(crys[(size_t)b * DD + t] / c);
    }
    __syncthreads();
    float acc = cfb[t];
    for (int d = 0; d < DD; ++d) acc += p[d] * cfw[d * HH + t];
    red[t] = softplusf(acc) * ow[t];
    __syncthreads();
    for (int off = 64; off > 0; off >>= 1) {
        if (t < off) red[t] += red[t + off];
        __syncthreads();
    }
    if (t == 0) out[b] = red[0] + ob[0];
}

// ---------------------------------------------------------------------------
extern "C" void kernel_launch(void* const* d_in, const int* in_sizes, int n_in,
                              void* d_out, int out_size, void* d_ws, size_t ws_size,
                              hipStream_t stream)
{
    const float* atom_fea = (const float*)d_in[0];
    const float* nbr_fea  = (const float*)d_in[1];
    const int*   nbr_idx  = (const int*)d_in[2];
    const int*   cids     = (const int*)d_in[3];
    const float* emb_w    = (const float*)d_in[4];
    const float* emb_b    = (const float*)d_in[5];
    const float* fc_w     = (const float*)d_in[6];
    const float* fc_b     = (const float*)d_in[7];
    const float* bn1_g    = (const float*)d_in[8];
    const float* bn1_b    = (const float*)d_in[9];
    const float* bn2_g    = (const float*)d_in[10];
    const float* bn2_b    = (const float*)d_in[11];
    const float* cf_w     = (const float*)d_in[12];
    const float* cf_b     = (const float*)d_in[13];
    const float* out_w    = (const float*)d_in[14];
    const float* out_b    = (const float*)d_in[15];
    float* out = (float*)d_out;

    float* ws    = (float*)d_ws;
    float* x     = ws;
    float* s     = x + (size_t)NN * DD;
    float* stat1 = s + (size_t)NN * DD;
    float* bnp1  = stat1 + 2 * TWO_D;
    float* stat2 = bnp1 + 2 * TWO_D;
    float* bnp2  = stat2 + 2 * DD;
    float* crys  = bnp2 + 2 * DD;
    float* cnt   = crys + (size_t)NCRYS * DD;
    size_t need = (size_t)(cnt + NCRYS - ws) * sizeof(float);
    if (ws_size < need) return;

    // 1. embedding
    embed_kernel<<<(NN + 63) / 64, 128, 0, stream>>>(atom_fea, emb_w, emb_b, x);

    // 2. conv layers
    const int nblk = (NN * MM) / 48;             // 15000
    for (int l = 0; l < 3; ++l) {
        const float* fw = fc_w + (size_t)l * KC * TWO_D;
        const float* fb = fc_b + (size_t)l * TWO_D;
        zero_f32<<<1, 256, 0, stream>>>(stat1, 2 * TWO_D);
        conv_pass1<<<nblk, 256, 0, stream>>>(x, nbr_fea, nbr_idx, fw, fb, stat1);
        bn_finalize<<<1, 128, 0, stream>>>(stat1, bnp1, TWO_D, 1.0f / (float)(NN * MM));
        zero_f32<<<1, 128, 0, stream>>>(stat2, 2 * DD);
        conv_pass2<<<nblk, 256, 0, stream>>>(x, nbr_fea, nbr_idx, fw, fb, bnp1,
                                             bn1_g + l * TWO_D, bn1_b + l * TWO_D,
                                             s, stat2);
        bn_finalize<<<1, 64, 0, stream>>>(stat2, bnp2, DD, 1.0f / (float)NN);
        update_x<<<(NN * DD) / 256, 256, 0, stream>>>(x, s, bnp2,
                                                      bn2_g + l * DD, bn2_b + l * DD);
    }

    // 3. crystal pooling
    zero_f32<<<(NCRYS * DD + NCRYS + 255) / 256, 256, 0, stream>>>(crys, NCRYS * DD + NCRYS);
    pool_kernel<<<(NN * DD) / 256, 256, 0, stream>>>(x, cids, crys, cnt);

    // 4. head
    head_kernel<<<NCRYS, 128, 0, stream>>>(crys, cnt, cf_w, cf_b, out_w, out_b, out);
}